// IGMC_33827162423506
// MI455X (gfx1250) — compile-verified
//
#include <hip/hip_runtime.h>

// CDNA5 / gfx1250: wave32, WMMA 16x16 tiles.
typedef __attribute__((ext_vector_type(2))) float v2f;
typedef __attribute__((ext_vector_type(8))) float v8f;

#define HIDF 64   // HID == OUT_DIM == 64 (fixed by the reference model)

// ---------------- degree / normalization ----------------
__global__ void k_init_deg(float* __restrict__ deg, int N) {
  int i = blockIdx.x * blockDim.x + threadIdx.x;
  if (i < N) deg[i] = 1.0f;                       // self-loop contribution
}

__global__ void k_deg_accum(const int* __restrict__ dst, float* __restrict__ deg, int E) {
  int e = blockIdx.x * blockDim.x + threadIdx.x;
  if (e < E) atomicAdd(&deg[dst[e]], 1.0f);       // global_atomic_add_f32, no-return
}

__global__ void k_rsqrt(float* __restrict__ deg, int N) {
  int i = blockIdx.x * blockDim.x + threadIdx.x;
  if (i < N) deg[i] = rsqrtf(deg[i]);             // deg >= 1 guaranteed
}

// ---------------- fp32 WMMA GEMM: T[M,64] = H[M,K] @ W[K,64] (+bias,relu) ----
// Block = 128 threads = 4 waves. Wave w computes the 16x16 tile at
// (m0 = 16*blockIdx.x, n0 = 16*w). K stepped by 4 with V_WMMA_F32_16X16X4_F32.
//
// fp32 A 16x4 layout (ISA 7.12.2): lanes 0-15 -> M=lane,  VGPR0=K0, VGPR1=K1
//                                  lanes 16-31 -> M=lane-16, VGPR0=K2, VGPR1=K3
// B 4x16 mirrors with N striped across lanes.
// C/D 16x16 f32: VGPR r: lanes0-15 -> M=r, lanes16-31 -> M=r+8, N=lane&15.
__global__ void k_gemm_wmma(const float* __restrict__ H, const float* __restrict__ W,
                            const float* __restrict__ bias, float* __restrict__ T,
                            int K, int fuseBiasRelu) {
  const int wave  = threadIdx.x >> 5;
  const int lane  = threadIdx.x & 31;
  const int lhalf = lane >> 4;      // 0: lanes 0-15, 1: lanes 16-31
  const int l16   = lane & 15;
  const int m0    = blockIdx.x * 16;
  const int n0    = wave * 16;      // HIDF==64 -> 4 n-tiles per block

  const float* __restrict__ arow = H + (size_t)(m0 + l16) * K;
  v8f acc = {};

  for (int k0 = 0; k0 < K; k0 += 4) {
    const int ka = k0 + 2 * lhalf;
    v2f a;
    a.x = arow[ka + 0];
    a.y = arow[ka + 1];
    v2f b;
    b.x = W[(size_t)(ka + 0) * HIDF + n0 + l16];
    b.y = W[(size_t)(ka + 1) * HIDF + n0 + l16];
    // (neg_a, A, neg_b, B, c_mod, C, reuse_a, reuse_b)
    acc = __builtin_amdgcn_wmma_f32_16x16x4_f32(false, a, false, b,
                                                (short)0, acc, false, false);
  }

  float* __restrict__ orow = T + (size_t)m0 * HIDF + n0 + l16;
#pragma unroll
  for (int r = 0; r < 8; ++r) {
    int row = r + 8 * lhalf;
    float v = acc[r];
    if (fuseBiasRelu) {
      v += bias[n0 + l16];
      v = fmaxf(v, 0.0f);
    }
    orow[(size_t)row * HIDF] = v;
  }
}

// ---------------- self-loop init: agg = T * dis^2 ----------------
// One thread per (node, float4-chunk); 16 chunks cover the 64-wide row.
__global__ void k_selfloop(const float* __restrict__ T, const float* __restrict__ dis,
                           float* __restrict__ agg, int N) {
  int i = blockIdx.x * blockDim.x + threadIdx.x;
  int n = i >> 4, q = i & 15;
  if (n >= N) return;
  float s = dis[n];
  s = s * s;
  float4 v = ((const float4*)(T + (size_t)n * HIDF))[q];
  float4 r = make_float4(v.x * s, v.y * s, v.z * s, v.w * s);
  ((float4*)(agg + (size_t)n * HIDF))[q] = r;
}

// ---------------- edge scatter: agg[dst] += T[src] * dis[src]*dis[dst] -----
// 16 lanes per edge, float4 gather (256B contiguous row) + 4 f32 atomics.
__global__ void k_scatter(const float* __restrict__ T, const int* __restrict__ src,
                          const int* __restrict__ dst, const float* __restrict__ dis,
                          float* __restrict__ agg, int E) {
  long long gid = (long long)blockIdx.x * blockDim.x + threadIdx.x;
  int e = (int)(gid >> 4);
  int q = (int)(gid & 15);
  if (e >= E) return;
  int s = src[e], d = dst[e];
  float nrm = dis[s] * dis[d];
  float4 v = ((const float4*)(T + (size_t)s * HIDF))[q];
  float* o = agg + (size_t)d * HIDF + 4 * q;
  atomicAdd(o + 0, v.x * nrm);
  atomicAdd(o + 1, v.y * nrm);
  atomicAdd(o + 2, v.z * nrm);
  atomicAdd(o + 3, v.w * nrm);
}

// ---------------- epilogue: h = relu(agg + b) (in place) ----------------
__global__ void k_bias_relu(float* __restrict__ agg, const float* __restrict__ b, int total) {
  int i = blockIdx.x * blockDim.x + threadIdx.x;
  if (i < total) {
    float v = agg[i] + b[i & (HIDF - 1)];
    agg[i] = fmaxf(v, 0.0f);
  }
}

// ---------------- host launcher ----------------
static inline int ceil_div(long long a, int b) { return (int)((a + b - 1) / b); }

extern "C" void kernel_launch(void* const* d_in, const int* in_sizes, int n_in,
                              void* d_out, int out_size, void* d_ws, size_t ws_size,
                              hipStream_t stream) {
  const float* x     = (const float*)d_in[0];
  const int*   ei    = (const int*)  d_in[1];
  const float* W1    = (const float*)d_in[2];
  const float* b1    = (const float*)d_in[3];
  const float* W2    = (const float*)d_in[4];
  const float* b2    = (const float*)d_in[5];
  const float* W3    = (const float*)d_in[6];
  const float* b3    = (const float*)d_in[7];
  const float* Wout  = (const float*)d_in[8];
  const float* bout  = (const float*)d_in[9];
  float* out = (float*)d_out;

  const int K1 = in_sizes[2] / HIDF;     // IN_DIM (128)
  const int N  = in_sizes[0] / K1;       // 100000 (multiple of 16)
  const int E  = in_sizes[1] / 2;        // 1.6M
  const int* src = ei;
  const int* dst = ei + E;

  // scratch layout in d_ws (256B aligned slices)
  char* ws = (char*)d_ws;
  size_t off = 0;
  auto take = [&](size_t bytes) -> void* {
    void* p = ws + off;
    off = (off + bytes + 255) & ~(size_t)255;
    return p;
  };
  float* dis = (float*)take((size_t)N * 4);
  float* T   = (float*)take((size_t)N * HIDF * 4);
  float* hA  = (float*)take((size_t)N * HIDF * 4);
  float* hB  = (float*)take((size_t)N * HIDF * 4);
  (void)ws_size; (void)n_in; (void)out_size;

  const int BT = 256;
  dim3 gemmGrid(N / 16), gemmBlk(128);

  // --- normalization ---
  k_init_deg <<<ceil_div(N, BT), BT, 0, stream>>>(dis, N);
  k_deg_accum<<<ceil_div(E, BT), BT, 0, stream>>>(dst, dis, E);
  k_rsqrt    <<<ceil_div(N, BT), BT, 0, stream>>>(dis, N);

  // --- layer 1: x[ N x K1 ] @ W1 -> T ; aggregate -> hA ; relu(hA+b1) ---
  k_gemm_wmma<<<gemmGrid, gemmBlk, 0, stream>>>(x, W1, b1, T, K1, 0);
  k_selfloop <<<ceil_div((long long)N * 16, BT), BT, 0, stream>>>(T, dis, hA, N);
  k_scatter  <<<ceil_div((long long)E * 16, BT), BT, 0, stream>>>(T, src, dst, dis, hA, E);
  k_bias_relu<<<ceil_div((long long)N * HIDF, BT), BT, 0, stream>>>(hA, b1, N * HIDF);

  // --- layer 2: hA @ W2 -> T ; aggregate -> hB ---
  k_gemm_wmma<<<gemmGrid, gemmBlk, 0, stream>>>(hA, W2, b2, T, HIDF, 0);
  k_selfloop <<<ceil_div((long long)N * 16, BT), BT, 0, stream>>>(T, dis, hB, N);
  k_scatter  <<<ceil_div((long long)E * 16, BT), BT, 0, stream>>>(T, src, dst, dis, hB, E);
  k_bias_relu<<<ceil_div((long long)N * HIDF, BT), BT, 0, stream>>>(hB, b2, N * HIDF);

  // --- layer 3: hB @ W3 -> T ; aggregate -> hA ---
  k_gemm_wmma<<<gemmGrid, gemmBlk, 0, stream>>>(hB, W3, b3, T, HIDF, 0);
  k_selfloop <<<ceil_div((long long)N * 16, BT), BT, 0, stream>>>(T, dis, hA, N);
  k_scatter  <<<ceil_div((long long)E * 16, BT), BT, 0, stream>>>(T, src, dst, dis, hA, E);
  k_bias_relu<<<ceil_div((long long)N * HIDF, BT), BT, 0, stream>>>(hA, b3, N * HIDF);

  // --- output: relu(hA @ Wout + bout) -> d_out (bias+relu fused in epilogue) ---
  k_gemm_wmma<<<gemmGrid, gemmBlk, 0, stream>>>(hA, Wout, bout, out, HIDF, 1);
}